// DotProductCostVolume_4939212390830
// MI455X (gfx1250) — compile-verified
//
#include <hip/hip_runtime.h>

typedef float v2f __attribute__((ext_vector_type(2)));
typedef float v8f __attribute__((ext_vector_type(8)));

#define B_  4
#define C_  32
#define H_  256
#define W_  512
#define D_  64
#define NT  5          // delta tiles 0..4 cover d in [0,64)

// One wave32 per (b, h, 16-wide w-tile).
//  Phase 1: 5 Gram tiles G[w, w'] = sum_c L[c,w] R[c,w'] via v_wmma_f32_16x16x4_f32,
//           A (L tile) fragments loaded once and reused across all 5 delta tiles.
//           B loads are UNCONDITIONAL (address clamped to 0); out-of-image columns
//           (w' < 0) are zeroed with a post-load select so loads stay branch-free
//           and can be clustered/overlapped with the WMMAs.
//  Phase 2: accumulators staged in LDS, re-read along diagonals d = w - w' so the
//           global stores are contiguous 16-float (64B) runs per disparity.
__global__ __launch_bounds__(32) void cost_volume_wmma(
    const float* __restrict__ L, const float* __restrict__ R, float* __restrict__ out)
{
    __shared__ float g[NT * 256];

    const int lane  = threadIdx.x;   // 0..31
    const int half  = lane >> 4;     // 0 or 1
    const int l16   = lane & 15;

    int blk = blockIdx.x;
    const int wt = blk & 31;         // W/16 = 32 tiles
    blk >>= 5;
    const int h  = blk & 255;        // H = 256
    const int b  = blk >> 8;
    const int w0 = wt * 16;

    const size_t planeHW = (size_t)H_ * W_;
    const float* Lbase = L + (size_t)b * C_ * planeHW + (size_t)h * W_;
    const float* Rbase = R + (size_t)b * C_ * planeHW + (size_t)h * W_;

    // ---- A fragments: 8 k-steps of 4 channels each (ISA 16x4 f32 A layout) ----
    // lane<16  : M=l16, K = 4k+0 (.x), 4k+1 (.y)
    // lane>=16 : M=l16, K = 4k+2 (.x), 4k+3 (.y)
    v2f afrag[8];
#pragma unroll
    for (int k = 0; k < 8; ++k) {
        const int c0 = 4 * k + 2 * half;
        const float* p = Lbase + (size_t)c0 * planeHW + (w0 + l16);
        afrag[k].x = p[0];
        afrag[k].y = p[planeHW];
    }

    // ---- 5 delta tiles, A reused, B address-clamped + value-zeroed for w' < 0 ----
#pragma unroll
    for (int delta = 0; delta < NT; ++delta) {
        const int wp_raw = w0 - 16 * delta + l16;     // B column w' for this lane
        const int wp     = (wp_raw < 0) ? 0 : wp_raw; // clamp: load always in-bounds
        const bool valid = (wp_raw >= 0);

        v8f acc = {};
#pragma unroll
        for (int k = 0; k < 8; ++k) {
            const int c0 = 4 * k + 2 * half;
            const float* p = Rbase + (size_t)c0 * planeHW + wp;
            const float x = p[0];          // unconditional loads -> clustered
            const float y = p[planeHW];
            v2f bfrag;
            bfrag.x = valid ? x : 0.0f;    // cheap v_cndmask, no branch
            bfrag.y = valid ? y : 0.0f;
            acc = __builtin_amdgcn_wmma_f32_16x16x4_f32(
                      /*neg_a=*/false, afrag[k],
                      /*neg_b=*/false, bfrag,
                      /*c_mod=*/(short)0, acc,
                      /*reuse_a=*/false, /*reuse_b=*/false);
        }

        // C/D layout: VGPR v -> (M = v + 8*half, N = l16)
#pragma unroll
        for (int v = 0; v < 8; ++v) {
            const int m = v + 8 * half;
            g[delta * 256 + m * 16 + l16] = acc[v];
        }
    }

    __syncthreads();   // single-wave WG: cheap; orders DS stores vs reads

    // ---- Phase 2: diagonal gather from LDS, coalesced stores ----
    // out[b, d, h, w0 + n], n = l16; lanes 0-15 do disparity d, lanes 16-31 do d+1.
    const float inv = 1.0f / (float)C_;
    float* obase = out + ((size_t)b * D_ * H_ + h) * W_ + w0;
#pragma unroll 4
    for (int it = 0; it < 32; ++it) {
        const int d  = 2 * it + half;
        const int n  = l16;
        const int dn = d - n;
        const int delta = (dn <= 0) ? 0 : ((dn + 15) >> 4);   // ceil(dn/16) in [0,4]
        const int np    = 16 * delta + n - d;                 // column within tile, 0..15
        // G entry (m=n, n'=np) of tile delta == G[w, w-d]; already 0 when w < d.
        const float val = g[delta * 256 + n * 16 + np] * inv;
        obase[(size_t)d * planeHW + n] = val;
    }
}

extern "C" void kernel_launch(void* const* d_in, const int* in_sizes, int n_in,
                              void* d_out, int out_size, void* d_ws, size_t ws_size,
                              hipStream_t stream) {
    (void)in_sizes; (void)n_in; (void)out_size; (void)d_ws; (void)ws_size;
    const float* L = (const float*)d_in[0];
    const float* R = (const float*)d_in[1];
    float* out = (float*)d_out;

    const int blocks = B_ * H_ * (W_ / 16);   // 4 * 256 * 32 = 32768
    hipLaunchKernelGGL(cost_volume_wmma, dim3(blocks), dim3(32), 0, stream, L, R, out);
}